// PairedKidneyCriticModel_91216515432551
// MI455X (gfx1250) — compile-verified
//
#include <hip/hip_runtime.h>
#include <hip/hip_bf16.h>
#include <math.h>

typedef float v2f __attribute__((ext_vector_type(2)));
typedef float v8f __attribute__((ext_vector_type(8)));

#define HDIM 64
#define NEG_SLOPE 0.2f
#define LN_EPS 1e-5f

// ---------------------------------------------------------------------------
// prep: fold the two embedding linears (no nonlinearity between them) into
// W12 (2 x 64) and b12 (64); zero the global accumulator.
// x = (feat @ w1 + b1) @ w2 + b2  ==  feat @ (w1@w2) + (b1@w2 + b2)
// ---------------------------------------------------------------------------
__global__ void prep_kernel(const float* __restrict__ emb_w1,  // [2,64]
                            const float* __restrict__ emb_b1,  // [64]
                            const float* __restrict__ emb_w2,  // [64,64]
                            const float* __restrict__ emb_b2,  // [64]
                            float* __restrict__ w12,           // [2*64]
                            float* __restrict__ b12,           // [64]
                            float* __restrict__ acc) {
  int j = threadIdx.x;
  if (j == 0) *acc = 0.f;
  if (j < HDIM) {
    float s0 = 0.f, s1 = 0.f, sb = 0.f;
    for (int k = 0; k < HDIM; ++k) {
      float w2 = emb_w2[k * HDIM + j];
      s0 += emb_w1[k] * w2;            // row 0 of emb_w1
      s1 += emb_w1[HDIM + k] * w2;     // row 1 of emb_w1
      sb += emb_b1[k] * w2;
    }
    w12[j]        = s0;
    w12[HDIM + j] = s1;
    b12[j]        = sb + emb_b2[j];
  }
}

// ---------------------------------------------------------------------------
// mask: mask[i][j] = (adj[j][i] != 0) | (i == j), packed 32 bits per word.
// Reads the 64MB adjacency exactly once, coalesced along i.
// ---------------------------------------------------------------------------
__global__ __launch_bounds__(256)
void mask_kernel(const float* __restrict__ adj, unsigned* __restrict__ maskW, int N) {
  int lane = threadIdx.x;                       // 0..31  -> i offset
  int i = blockIdx.x * 32 + lane;
  int w = blockIdx.y * blockDim.y + threadIdx.y;
  int jbase = w * 32;
  unsigned bits = 0u;
  for (int k = 0; k < 32; ++k) {
    float v = adj[(size_t)(jbase + k) * N + i]; // coalesced across lanes
    bits |= (v != 0.f ? 1u : 0u) << k;
  }
  if (i >= jbase && i < jbase + 32) bits |= 1u << (i - jbase);  // self-loop
  maskW[(size_t)i * (N >> 5) + w] = bits;
}

// ---------------------------------------------------------------------------
// features: x0[i][j] = progress*W12[0][j] + hard*W12[1][j] + b12[j]
// ---------------------------------------------------------------------------
__global__ __launch_bounds__(256)
void features_kernel(const int* __restrict__ timestep,
                     const float* __restrict__ arrivals,
                     const float* __restrict__ departures,
                     const float* __restrict__ hard,
                     const float* __restrict__ w12,
                     const float* __restrict__ b12,
                     float* __restrict__ x0, int N) {
  int idx = blockIdx.x * blockDim.x + threadIdx.x;
  if (idx >= N * HDIM) return;
  int i = idx >> 6, j = idx & 63;
  float t = (float)timestep[0];
  float p = (t - arrivals[i]) / (departures[i] - arrivals[i]);
  x0[idx] = p * w12[j] + hard[i] * w12[HDIM + j] + b12[j];
}

// ---------------------------------------------------------------------------
// hw = A (N x 64) @ W (64 x 64), fp32 via V_WMMA_F32_16X16X4_F32.
// One wave per 16x16 output tile; 4 waves/block cover the 4 column tiles.
// Fragment layouts per ISA 7.12.2 (A 16x4, B 4x16, C/D 16x16 f32).
// EXEC is all ones (no divergence, exact tiling).
// ---------------------------------------------------------------------------
__global__ __launch_bounds__(128)
void gemm_wmma_kernel(const float* __restrict__ A, const float* __restrict__ W,
                      float* __restrict__ C) {
  const int lane = threadIdx.x & 31;
  const int i0 = blockIdx.x * 16;        // row tile
  const int j0 = (int)threadIdx.y * 16;  // column tile (0,16,32,48)
  const int half = lane >> 4;            // 0: lanes 0-15, 1: lanes 16-31
  const int l16  = lane & 15;

  v8f acc = {};
  for (int k = 0; k < HDIM; k += 4) {
    // A frag: M = l16; K = k+2*half (+0,+1)
    const float* arow = A + (size_t)(i0 + l16) * HDIM + k + half * 2;
    v2f a; a.x = arow[0]; a.y = arow[1];
    // B frag: N = l16; K = k+2*half (VGPR0), k+2*half+1 (VGPR1)
    const float* bcol = W + (size_t)(k + half * 2) * HDIM + j0 + l16;
    v2f b; b.x = bcol[0]; b.y = bcol[HDIM];
    acc = __builtin_amdgcn_wmma_f32_16x16x4_f32(
        /*neg_a=*/false, a, /*neg_b=*/false, b,
        /*c_mod=*/(short)0, acc, /*reuse_a=*/false, /*reuse_b=*/false);
  }
  // D: VGPR r -> M = r + 8*half, N = l16
  #pragma unroll
  for (int r = 0; r < 8; ++r)
    C[(size_t)(i0 + half * 8 + r) * HDIM + j0 + l16] = acc[r];
}

// ---------------------------------------------------------------------------
// scores: ssrc[i] = hw[i].asrc ; sdst[i] = hw[i].adst
// ---------------------------------------------------------------------------
__global__ __launch_bounds__(256)
void scores_kernel(const float* __restrict__ hw,
                   const float* __restrict__ asrc, const float* __restrict__ adst,
                   float* __restrict__ ssrc, float* __restrict__ sdst, int N) {
  int i = blockIdx.x * blockDim.x + threadIdx.x;
  if (i >= N) return;
  float a = 0.f, b = 0.f;
  const float* row = hw + (size_t)i * HDIM;
  for (int k = 0; k < HDIM; ++k) { float v = row[k]; a += v * asrc[k]; b += v * adst[k]; }
  ssrc[i] = a; sdst[i] = b;
}

// ---------------------------------------------------------------------------
// attention: one wave32 per destination row i.  Sparse gather over the
// bitmask with an online (rescaling) softmax; lane l owns features l, l+32.
// h[i] = softmax_j(lrelu(ssrc[j]+sdst[i])) @ hw + bias  [+ReLU between layers]
// ---------------------------------------------------------------------------
__global__ __launch_bounds__(256)
void attn_kernel(const unsigned* __restrict__ maskW, const float* __restrict__ hw,
                 const float* __restrict__ ssrc, const float* __restrict__ sdst,
                 const float* __restrict__ bias, int do_relu,
                 float* __restrict__ hout, int N) {
  int wave = threadIdx.x >> 5;
  int lane = threadIdx.x & 31;
  int i = blockIdx.x * 8 + wave;
  if (i >= N) return;

  const int nwords = N >> 5;
  const unsigned* mrow = maskW + (size_t)i * nwords;
  float sd = sdst[i];
  float m = -3.0e38f, sumw = 0.f, a0 = 0.f, a1 = 0.f;

  for (int w = 0; w < nwords; ++w) {
    unsigned bits = mrow[w];                 // uniform across the wave
    while (bits) {
      int b = __ffs(bits) - 1;
      bits &= bits - 1;
      int j = (w << 5) + b;
      float e = ssrc[j] + sd;
      e = (e > 0.f) ? e : NEG_SLOPE * e;     // leaky_relu
      float wgt;
      if (e > m) {                           // online softmax rescale
        float r = __expf(m - e);
        sumw *= r; a0 *= r; a1 *= r;
        m = e; wgt = 1.f;
      } else {
        wgt = __expf(e - m);
      }
      sumw += wgt;
      const float* hrow = hw + (size_t)j * HDIM;
      a0 += wgt * hrow[lane];
      a1 += wgt * hrow[lane + 32];
    }
  }
  float inv = 1.f / sumw;                    // self-loop guarantees sumw > 0
  float h0 = a0 * inv + bias[lane];
  float h1 = a1 * inv + bias[lane + 32];
  if (do_relu) { h0 = h0 > 0.f ? h0 : 0.f; h1 = h1 > 0.f ? h1 : 0.f; }
  hout[(size_t)i * HDIM + lane]      = h0;
  hout[(size_t)i * HDIM + lane + 32] = h1;
}

// ---------------------------------------------------------------------------
// finalize: per-row residual + layernorm, dot with val_w, reduce into acc.
// One wave32 per row; __shfl_xor tree reductions (wave32 width).
// ---------------------------------------------------------------------------
__global__ __launch_bounds__(256)
void finalize_kernel(const float* __restrict__ x0, const float* __restrict__ h,
                     const float* __restrict__ valw, float* __restrict__ acc, int N) {
  int wave = threadIdx.x >> 5;
  int lane = threadIdx.x & 31;
  int i = blockIdx.x * 8 + wave;
  if (i >= N) return;
  size_t base = (size_t)i * HDIM;
  float v0 = x0[base + lane]      + h[base + lane];
  float v1 = x0[base + lane + 32] + h[base + lane + 32];

  float s = v0 + v1;
  for (int off = 16; off; off >>= 1) s += __shfl_xor(s, off, 32);
  float mu = s * (1.f / (float)HDIM);
  float d0 = v0 - mu, d1 = v1 - mu;
  float q = d0 * d0 + d1 * d1;
  for (int off = 16; off; off >>= 1) q += __shfl_xor(q, off, 32);
  float rstd = rsqrtf(q * (1.f / (float)HDIM) + LN_EPS);

  float dot = (d0 * rstd) * valw[lane] + (d1 * rstd) * valw[lane + 32];
  for (int off = 16; off; off >>= 1) dot += __shfl_xor(dot, off, 32);
  if (lane == 0) atomicAdd(acc, dot);
}

__global__ void epilogue_kernel(const float* __restrict__ acc,
                                const float* __restrict__ valb,
                                float* __restrict__ out, int N) {
  float v = acc[0] / (float)N + valb[0];
  out[0] = v > 0.f ? v : 0.f;
}

// ---------------------------------------------------------------------------
// launch
// ---------------------------------------------------------------------------
extern "C" void kernel_launch(void* const* d_in, const int* in_sizes, int n_in,
                              void* d_out, int out_size, void* d_ws, size_t ws_size,
                              hipStream_t stream) {
  const float* adj      = (const float*)d_in[0];
  const int*   timestep = (const int*)  d_in[1];
  const float* arrivals = (const float*)d_in[2];
  const float* depart   = (const float*)d_in[3];
  const float* hard     = (const float*)d_in[4];
  // d_in[5] = active_agents (unused by the reference math)
  const float* emb_w1   = (const float*)d_in[6];
  const float* emb_b1   = (const float*)d_in[7];
  const float* emb_w2   = (const float*)d_in[8];
  const float* emb_b2   = (const float*)d_in[9];
  const float* gat_w    = (const float*)d_in[10];  // [3,64,64]
  const float* gat_asrc = (const float*)d_in[11];  // [3,64]
  const float* gat_adst = (const float*)d_in[12];  // [3,64]
  const float* gat_b    = (const float*)d_in[13];  // [3,64]
  const float* val_w    = (const float*)d_in[14];  // [64]
  const float* val_b    = (const float*)d_in[15];  // [1]
  float* out = (float*)d_out;

  const int N = in_sizes[2];      // 4096
  const int L = 3;

  // workspace layout (bytes)
  char* ws = (char*)d_ws;
  size_t off = 0;
  unsigned* maskW = (unsigned*)(ws + off); off += (size_t)N * (N >> 5) * 4;  // 2 MB
  float* x0   = (float*)(ws + off); off += (size_t)N * HDIM * 4;            // 1 MB
  float* h    = (float*)(ws + off); off += (size_t)N * HDIM * 4;            // 1 MB
  float* hwb  = (float*)(ws + off); off += (size_t)N * HDIM * 4;            // 1 MB
  float* ssrc = (float*)(ws + off); off += (size_t)N * 4;
  float* sdst = (float*)(ws + off); off += (size_t)N * 4;
  float* w12  = (float*)(ws + off); off += 2 * HDIM * 4;
  float* b12  = (float*)(ws + off); off += HDIM * 4;
  float* acc  = (float*)(ws + off); off += 64;  // padded
  (void)ws_size; (void)n_in; (void)out_size;

  // 1) fold embedding weights + zero accumulator
  prep_kernel<<<1, 64, 0, stream>>>(emb_w1, emb_b1, emb_w2, emb_b2, w12, b12, acc);

  // 2) adjacency -> bitmask (single pass over the 64MB matrix)
  {
    dim3 blk(32, 8), grd(N / 32, (N >> 5) / 8);
    mask_kernel<<<grd, blk, 0, stream>>>(adj, maskW, N);
  }

  // 3) node features
  features_kernel<<<(N * HDIM + 255) / 256, 256, 0, stream>>>(
      timestep, arrivals, depart, hard, w12, b12, x0, N);

  // 4) GAT layers
  for (int l = 0; l < L; ++l) {
    const float* hin = (l == 0) ? x0 : h;
    dim3 gblk(32, 4);
    gemm_wmma_kernel<<<N / 16, gblk, 0, stream>>>(hin, gat_w + (size_t)l * HDIM * HDIM, hwb);
    scores_kernel<<<(N + 255) / 256, 256, 0, stream>>>(
        hwb, gat_asrc + (size_t)l * HDIM, gat_adst + (size_t)l * HDIM, ssrc, sdst, N);
    attn_kernel<<<N / 8, 256, 0, stream>>>(
        maskW, hwb, ssrc, sdst, gat_b + (size_t)l * HDIM, (l < L - 1) ? 1 : 0, h, N);
  }

  // 5) residual + layernorm + value head reduction
  finalize_kernel<<<N / 8, 256, 0, stream>>>(x0, h, val_w, acc, N);
  epilogue_kernel<<<1, 1, 0, stream>>>(acc, val_b, out, N);
}